// GCN_34205119545813
// MI455X (gfx1250) — compile-verified
//
#include <hip/hip_runtime.h>
#include <hip/hip_bf16.h>

// ---------------------------------------------------------------------------
// GCN pipeline for MI455X (gfx1250, wave32).
// Dominant cost: 6x dense A[8192x8192] @ H[8192,k] fp32 matmuls (k = 1 or 4)
// -> pure HBM-bandwidth problem (<=2 FLOP/byte). A is streamed with
// GLOBAL_LOAD_ASYNC_TO_LDS_B128 (ASYNCcnt, no VGPR round-trip) and consumed
// by V_WMMA_F32_16X16X4_F32 via conflict-free ds_load_b64.
// No fp atomics anywhere -> bitwise-deterministic across graph replays.
// ---------------------------------------------------------------------------

typedef float v2f __attribute__((ext_vector_type(2)));
typedef float v4f __attribute__((ext_vector_type(4)));
typedef float v8f __attribute__((ext_vector_type(8)));

#define KSPLIT 8     // K-dimension split across blockIdx.y (disjoint partials)
#define ROWS   128   // output rows per block (8 waves x 16-row WMMA tiles)
#define KCH    64    // K columns staged in LDS per chunk
#define LPAD   68    // LDS row stride (dwords): 64 + 4 pad -> conflict-free ds_load_b64

#define USE_ASYNC_LDS 1   // flip to 0 if the async mnemonic fails to assemble

// Generic pointer to an LDS object: low 32 bits are the LDS byte offset
// (flat aperture rule).
__device__ __forceinline__ uint32_t lds_off(const void* p) {
  return (uint32_t)(uintptr_t)p;
}

// ---------------------------------------------------------------------------
// Y_partial[ks] = A[:, Kslice(ks)] @ Ht^T[Kslice(ks), :kc]
// Ht is stored TRANSPOSED: Ht[j*N + k] = H[k][j]  (contiguous along K).
// ---------------------------------------------------------------------------
__global__ __launch_bounds__(256) void bigK(const float* __restrict__ A,
                                            const float* __restrict__ Ht,
                                            float* __restrict__ Yp,
                                            int N, int kc) {
  __shared__ float As[ROWS * LPAD];

  const int tid  = threadIdx.x;
  const int lane = tid & 31;
  const int wave = tid >> 5;
  const int hi   = lane >> 4;     // upper/lower half of wave
  const int l16  = lane & 15;     // A-operand: row-in-tile; B/D: column
  const int R0   = blockIdx.x * ROWS;
  const int ks   = blockIdx.y;

  const int C  = N / KCH;                      // total K chunks (N % 64 == 0)
  const int c0 = (ks * C) / KSPLIT;
  const int c1 = ((ks + 1) * C) / KSPLIT;

  // Per-thread staging slots: 8 x (row r, float4 column c4). Row index is
  // CLAMPED (not branched) so every load is legal; the final store guard
  // keeps correctness for N % ROWS != 0.
  const float* gp[8];
  uint32_t     lo[8];
#pragma unroll
  for (int i = 0; i < 8; ++i) {
    int idx  = tid + 256 * i;     // 0..2047
    int r    = idx >> 4;          // 0..127
    int c4   = idx & 15;          // 0..15 float4s within 64-col chunk
    int grow = R0 + r;
    if (grow > N - 1) grow = N - 1;
    gp[i] = A + (size_t)grow * (size_t)N + (size_t)(4 * c4);
    lo[i] = lds_off(&As[r * LPAD + 4 * c4]);
  }

  // B operand: column l16 of the 16-wide tile reads H row (l16 mod kc).
  // Columns >= kc of D are never stored, so replicated (finite) data is fine
  // and we avoid a branch/select per WMMA.
  const float* hp = Ht + (size_t)(l16 & (kc - 1)) * (size_t)N + (size_t)(2 * hi);

  v8f acc = {};
  const float* ap = &As[(wave * 16 + l16) * LPAD + 2 * hi];

  for (int c = c0; c < c1; ++c) {
    const int k0 = c * KCH;
    __syncthreads();  // all waves done reading LDS from previous chunk
#if USE_ASYNC_LDS
    // 8 async b128 copies global->LDS per thread, all in flight at once.
#pragma unroll
    for (int i = 0; i < 8; ++i) {
      asm volatile("global_load_async_to_lds_b128 %0, %1, off"
                   :: "v"(lo[i]), "v"(gp[i] + k0)
                   : "memory");
    }
    asm volatile("s_wait_asynccnt 0" ::: "memory");
#else
    // Fallback: issue all 8 b128 loads, then all 8 ds_stores (keeps MLP).
    v4f stg[8];
#pragma unroll
    for (int i = 0; i < 8; ++i) stg[i] = *(const v4f*)(gp[i] + k0);
#pragma unroll
    for (int i = 0; i < 8; ++i) {
      int idx = tid + 256 * i, r = idx >> 4, c4 = idx & 15;
      *(v4f*)(&As[r * LPAD + 4 * c4]) = stg[i];
    }
#endif
    __syncthreads();

    // 16 WMMA steps of K=4. A-operand layout (ISA 7.12.2, 32-bit 16x4):
    //   lanes 0-15 & 16-31 both hold rows M=0..15; K = vgpr + 2*hi.
#pragma unroll
    for (int s = 0; s < 16; ++s) {
      v2f a = *(const v2f*)(ap + 4 * s);
      v2f b = *(const v2f*)(hp + k0 + 4 * s);
      acc = __builtin_amdgcn_wmma_f32_16x16x4_f32(false, a, false, b,
                                                  (short)0, acc, false, false);
    }
  }

  // D layout: N = l16, M = v + 8*hi. Plain stores into this ks's partial.
  if (l16 < kc) {
    float* dst = Yp + (size_t)ks * (size_t)N * (size_t)kc;
#pragma unroll
    for (int v = 0; v < 8; ++v) {
      int row = R0 + wave * 16 + v + 8 * hi;
      if (row < N) dst[(size_t)row * kc + l16] = acc[v];
    }
  }
}

// ---------------------------------------------------------------------------
// Yr = relu(sum_ks Yp[ks] + bias); per-block partial (sum, sumsq) for BN.
// Fixed 64-block grid + tree reduce -> deterministic.
// ---------------------------------------------------------------------------
__global__ __launch_bounds__(256) void statsK(const float* __restrict__ Yp,
                                              const float* __restrict__ bias,
                                              float* __restrict__ Yr,
                                              float* __restrict__ part,
                                              int elems, int kc, size_t stride) {
  float s = 0.f, sq = 0.f;
  for (int e = blockIdx.x * blockDim.x + threadIdx.x; e < elems;
       e += gridDim.x * blockDim.x) {
    float acc = bias[e % kc];
#pragma unroll
    for (int ks = 0; ks < KSPLIT; ++ks) acc += Yp[(size_t)ks * stride + e];
    acc = fmaxf(acc, 0.f);
    Yr[e] = acc;
    s += acc;
    sq += acc * acc;
  }
  __shared__ float rs[256], rq[256];
  int t = threadIdx.x;
  rs[t] = s; rq[t] = sq;
  __syncthreads();
  for (int o = 128; o > 0; o >>= 1) {
    if (t < o) { rs[t] += rs[t + o]; rq[t] += rq[t + o]; }
    __syncthreads();
  }
  if (t == 0) { part[blockIdx.x * 2] = rs[0]; part[blockIdx.x * 2 + 1] = rq[0]; }
}

__global__ __launch_bounds__(64) void finalizeK(const float* __restrict__ part,
                                                const float* __restrict__ gamma,
                                                const float* __restrict__ beta,
                                                float* __restrict__ sc,
                                                float inv_cnt) {
  __shared__ float rs[64], rq[64];
  int t = threadIdx.x;
  rs[t] = part[t * 2]; rq[t] = part[t * 2 + 1];
  __syncthreads();
  for (int o = 32; o > 0; o >>= 1) {
    if (t < o) { rs[t] += rs[t + o]; rq[t] += rq[t + o]; }
    __syncthreads();
  }
  if (t == 0) {
    float m = rs[0] * inv_cnt;
    float v = rq[0] * inv_cnt - m * m;
    float scale = gamma[0] * rsqrtf(v + 1e-5f);
    sc[0] = scale;
    sc[1] = beta[0] - m * scale;
  }
}

// --- tiny elementwise kernels ----------------------------------------------
// Ht[j][i] = x[i] * w[0][j]   (first projection of each branch)
__global__ void initHx(const float* __restrict__ x, const float* __restrict__ w,
                       float* __restrict__ Ht, int N) {
  int i = blockIdx.x * blockDim.x + threadIdx.x;
  if (i < N) {
    float xv = x[i];
#pragma unroll
    for (int j = 0; j < 4; ++j) Ht[(size_t)j * N + i] = xv * w[j];
  }
}

// Ht[0][i] = sum_k BN(Yr4[i][k]) * W[k]   (block post_relu = False -> identity)
__global__ void makeH41(const float* __restrict__ Yr4, const float* __restrict__ sc,
                        const float* __restrict__ W, float* __restrict__ Ht, int N) {
  int i = blockIdx.x * blockDim.x + threadIdx.x;
  if (i < N) {
    float s = sc[0], h = sc[1], a = 0.f;
#pragma unroll
    for (int k = 0; k < 4; ++k) a += (s * Yr4[(size_t)i * 4 + k] + h) * W[k];
    Ht[i] = a;
  }
}

// F[i] = relu(BN(Yr1[i]))   (block post_relu = True)
__global__ void makeFK(const float* __restrict__ Yr1, const float* __restrict__ sc,
                       float* __restrict__ F, int N) {
  int i = blockIdx.x * blockDim.x + threadIdx.x;
  if (i < N) F[i] = fmaxf(sc[0] * Yr1[i] + sc[1], 0.f);
}

// Ht[j][i] = F1[i]*wc1[0][j] + F2[i]*wc1[1][j]   (concat + projection)
__global__ void makeHcK(const float* __restrict__ F1, const float* __restrict__ F2,
                        const float* __restrict__ wc1, float* __restrict__ Ht, int N) {
  int i = blockIdx.x * blockDim.x + threadIdx.x;
  if (i < N) {
    float a = F1[i], b = F2[i];
#pragma unroll
    for (int j = 0; j < 4; ++j) Ht[(size_t)j * N + i] = a * wc1[j] + b * wc1[4 + j];
  }
}

__global__ void writeOutK(const float* __restrict__ Yr1, const float* __restrict__ sc,
                          float* __restrict__ out, int N) {
  int i = blockIdx.x * blockDim.x + threadIdx.x;
  if (i < N) out[i] = fmaxf(sc[0] * Yr1[i] + sc[1], 0.f);
}

// ---------------------------------------------------------------------------
extern "C" void kernel_launch(void* const* d_in, const int* in_sizes, int n_in,
                              void* d_out, int out_size, void* d_ws, size_t ws_size,
                              hipStream_t stream) {
  const float* x     = (const float*)d_in[0];   // [2,N,1]
  const float* adj   = (const float*)d_in[1];   // [3,N,N]
  const float* w11   = (const float*)d_in[2];
  const float* b11   = (const float*)d_in[3];
  const float* w12   = (const float*)d_in[4];
  const float* b12   = (const float*)d_in[5];
  const float* w21   = (const float*)d_in[6];
  const float* b21   = (const float*)d_in[7];
  const float* w22   = (const float*)d_in[8];
  const float* b22   = (const float*)d_in[9];
  const float* wc1   = (const float*)d_in[10];
  const float* bc1   = (const float*)d_in[11];
  const float* wc2   = (const float*)d_in[12];
  const float* bc2   = (const float*)d_in[13];
  const float* gamma = (const float*)d_in[14];
  const float* beta  = (const float*)d_in[15];

  const int    N  = in_sizes[0] / 2;
  const size_t NN = (size_t)N * (size_t)N;

  // workspace partition (fp32, all 16B aligned): ~43N floats total
  float* ws   = (float*)d_ws;
  float* Ht   = ws;                        // 4N  (transposed H, rows contiguous in K)
  float* Yp   = Ht + 4 * (size_t)N;        // KSPLIT * 4N partials
  float* Yr4  = Yp + (size_t)KSPLIT * 4 * (size_t)N;  // 4N
  float* Yr1  = Yr4 + 4 * (size_t)N;       // N
  float* Fa   = Yr1 + (size_t)N;           // N
  float* Fb   = Fa + (size_t)N;            // N
  float* part = Fb + (size_t)N;            // 128
  float* sc   = part + 128;                // 2

  dim3 bgrid((N + ROWS - 1) / ROWS, KSPLIT);
  int  eb = (N + 255) / 256;

  auto layer = [&](const float* A, const float* bias, int kc, float* Yr) {
    bigK<<<bgrid, 256, 0, stream>>>(A, Ht, Yp, N, kc);
    statsK<<<64, 256, 0, stream>>>(Yp, bias, Yr, part, N * kc, kc,
                                   (size_t)N * (size_t)kc);
    finalizeK<<<1, 64, 0, stream>>>(part, gamma, beta, sc,
                                    1.0f / (float)(N * kc));
  };

  // --- branch 1 (A1) ---
  initHx<<<eb, 256, 0, stream>>>(x, w11, Ht, N);
  layer(adj, b11, 4, Yr4);
  makeH41<<<eb, 256, 0, stream>>>(Yr4, sc, w12, Ht, N);
  layer(adj, b12, 1, Yr1);
  makeFK<<<eb, 256, 0, stream>>>(Yr1, sc, Fa, N);

  // --- branch 2 (A2) ---
  initHx<<<eb, 256, 0, stream>>>(x + N, w21, Ht, N);
  layer(adj + NN, b21, 4, Yr4);
  makeH41<<<eb, 256, 0, stream>>>(Yr4, sc, w22, Ht, N);
  layer(adj + NN, b22, 1, Yr1);
  makeFK<<<eb, 256, 0, stream>>>(Yr1, sc, Fb, N);

  // --- combined (A12) ---
  makeHcK<<<eb, 256, 0, stream>>>(Fa, Fb, wc1, Ht, N);
  layer(adj + 2 * NN, bc1, 4, Yr4);
  makeH41<<<eb, 256, 0, stream>>>(Yr4, sc, wc2, Ht, N);
  layer(adj + 2 * NN, bc2, 1, Yr1);
  writeOutK<<<eb, 256, 0, stream>>>(Yr1, sc, (float*)d_out, N);
}